// borderMask_17489106829544
// MI455X (gfx1250) — compile-verified
//
#include <hip/hip_runtime.h>

#ifndef __has_builtin
#define __has_builtin(x) 0
#endif

namespace {
constexpr int kN = 8;
constexpr int kH = 64;
constexpr int kW = 2048;
constexpr int kR = 3;                     // BORDER_SIZE erosion iterations == L1 radius
constexpr int kTileW = 64;
constexpr int kTileH = 16;
constexpr int kHaloW = kTileW + 2 * kR;   // 70
constexpr int kHaloH = kTileH + 2 * kR;   // 22
constexpr int kStride = 72;               // padded LDS row stride (dwords)
constexpr int kThreads = 256;             // 8 wave32 waves
}

// CDNA5 async global->LDS path (ASYNCcnt-tracked), with compile-safe fallback.
#if defined(__HIP_DEVICE_COMPILE__) && defined(__gfx1250__) && \
    __has_builtin(__builtin_amdgcn_global_load_async_to_lds_b32)
#define USE_ASYNC_LDS 1
#else
#define USE_ASYNC_LDS 0
#endif

#if USE_ASYNC_LDS
typedef __attribute__((address_space(1))) int* gptr_i32;
typedef __attribute__((address_space(3))) int* lptr_i32;
#endif

__global__ __launch_bounds__(kThreads)
void border_mask_kernel(const int* __restrict__ labels, float* __restrict__ out) {
    __shared__ int tile[kHaloH * kStride];

    const int bx = blockIdx.x;            // W / kTileW = 32
    const int by = blockIdx.y;            // H / kTileH = 4
    const int n  = blockIdx.z;            // N = 8
    const int x0 = bx * kTileW;
    const int y0 = by * kTileH;
    const int* __restrict__ img = labels + n * (kH * kW);
    const int tid = threadIdx.x;

    // ---- Stage halo tile into LDS (coordinate-clamped => always-valid addresses).
    for (int i = tid; i < kHaloH * kHaloW; i += kThreads) {
        const int ly = i / kHaloW;
        const int lx = i - ly * kHaloW;
        const int cy = min(max(y0 + ly - kR, 0), kH - 1);
        const int cx = min(max(x0 + lx - kR, 0), kW - 1);
        const int* src = img + cy * kW + cx;
        int* dst = &tile[ly * kStride + lx];
#if USE_ASYNC_LDS
        __builtin_amdgcn_global_load_async_to_lds_b32(
            (gptr_i32)src, (lptr_i32)dst, /*offset=*/0, /*cpol=*/0);
#else
        *dst = *src;
#endif
    }

#if defined(__gfx1250__) && defined(__HIP_DEVICE_COMPILE__)
#if USE_ASYNC_LDS && __has_builtin(__builtin_amdgcn_s_wait_asynccnt)
    __builtin_amdgcn_s_wait_asynccnt(0);
#else
    asm volatile("s_wait_asynccnt 0" ::: "memory");  // harmless no-op if no async ops issued
#endif
#endif
    __syncthreads();

    // ---- Compute: radius-3 L1 diamond equality stencil, 4 pixels per thread.
    const int tx  = tid & (kTileW - 1);   // 0..63
    const int tyq = tid >> 6;             // 0..3
    float* __restrict__ obase = out + (size_t)n * kH * kW;

#pragma unroll
    for (int k = 0; k < 4; ++k) {
        const int row = tyq * 4 + k;      // 0..15 within tile
        const int gy  = y0 + row;
        const int gx  = x0 + tx;
        const int center = tile[(kR + row) * kStride + (kR + tx)];
        int ok = 1;
#pragma unroll
        for (int dy = -kR; dy <= kR; ++dy) {
            const int w = kR - ((dy < 0) ? -dy : dy);
            const int yok = (unsigned)(gy + dy) < (unsigned)kH;
            const int base = (kR + row + dy) * kStride + (kR + tx);
#pragma unroll
            for (int dx = -kR; dx <= kR; ++dx) {
                if (dx < -w || dx > w) continue;       // outside diamond: dead-coded
                if (dy == 0 && dx == 0) continue;      // center trivially matches
                const int xok = (unsigned)(gx + dx) < (unsigned)kW;
                ok &= yok & xok & (tile[base + dx] == center);
            }
        }
        obase[gy * kW + gx] = ok ? 0.0f : 1.0f;        // border = NOT uniform diamond
    }
}

extern "C" void kernel_launch(void* const* d_in, const int* in_sizes, int n_in,
                              void* d_out, int out_size, void* d_ws, size_t ws_size,
                              hipStream_t stream) {
    (void)in_sizes; (void)n_in; (void)out_size; (void)d_ws; (void)ws_size;
    const int* labels = (const int*)d_in[0];
    float* out = (float*)d_out;
    dim3 grid(kW / kTileW, kH / kTileH, kN);   // 32 x 4 x 8 = 1024 blocks
    border_mask_kernel<<<grid, kThreads, 0, stream>>>(labels, out);
}